// RnCLoss_v3_79147657331273
// MI455X (gfx1250) — compile-verified
//
#include <hip/hip_runtime.h>
#include <hip/hip_bf16.h>
#include <math.h>

typedef float v2f __attribute__((ext_vector_type(2)));
typedef float v8f __attribute__((ext_vector_type(8)));

#define N 4096
#define DIMS 128
// TEMPERATURE = 2.0 -> logits = -dist * 0.5

// Workspace layout (in floats):
//   [0, 4096)            x2 row norms
//   [4096, 12288)        stage-1 per-block logit-sum partials (8192 slots)
//   [12288, 20480)       stage-2 per-block log-sum partials (2*4096 slots)
//   [32768, 32768+N*N)   E = exp(logits) matrix (64 MB)
#define WS_X2     0
#define WS_LOGITP 4096
#define WS_LOGP   12288
#define WS_E      32768

// ---------------------------------------------------------------- row norms
__global__ __launch_bounds__(128) void rnc_rownorm(const float* __restrict__ F,
                                                   float* __restrict__ x2) {
  __shared__ float red[128];
  const int row = blockIdx.x;
  const float v = F[row * DIMS + threadIdx.x];
  red[threadIdx.x] = v * v;
  __syncthreads();
  for (int s = 64; s > 0; s >>= 1) {
    if (threadIdx.x < s) red[threadIdx.x] += red[threadIdx.x + s];
    __syncthreads();
  }
  if (threadIdx.x == 0) x2[row] = red[0];
}

// ------------------------------------------- Gram via fp32 WMMA -> logits/E
// One wave computes one 16x16 tile of G = F*F^T with V_WMMA_F32_16X16X4_F32,
// then converts to logits = -sqrt(max(x2_i + x2_j - 2G, 0))/2, E = exp(logits),
// accumulating the block's sum of logits (diagonal forced to 0 exactly).
__global__ __launch_bounds__(256) void rnc_gram(const float* __restrict__ F,
                                                const float* __restrict__ x2,
                                                float* __restrict__ E,
                                                float* __restrict__ logitPart) {
  const int lane  = threadIdx.x & 31;
  const int wave  = threadIdx.x >> 5;
  const int itile = blockIdx.y * 16;
  const int jtile = (blockIdx.x * 8 + wave) * 16;

  const int r    = lane & 15;
  const int koff = (lane >> 4) << 1;  // lanes 0-15: K+0/K+1, lanes 16-31: K+2/K+3
  const float* arow = F + (itile + r) * DIMS + koff;
  const float* brow = F + (jtile + r) * DIMS + koff;

  v8f c = {};
#pragma unroll
  for (int k = 0; k < DIMS; k += 4) {
    v2f a = *(const v2f*)(arow + k);
    v2f b = *(const v2f*)(brow + k);
    // D = A(16x4) * B(4x16) + C   (fp32 WMMA, wave32)
    c = __builtin_amdgcn_wmma_f32_16x16x4_f32(false, a, false, b,
                                              (short)0, c, false, false);
  }

  // C layout: VGPR v -> M = v (lanes 0-15) or v+8 (lanes 16-31), N = lane&15
  const int  mbase = (lane >> 4) << 3;
  const int  gj    = jtile + r;
  const float x2j  = x2[gj];
  float lsum = 0.0f;
#pragma unroll
  for (int v = 0; v < 8; ++v) {
    const int gi = itile + mbase + v;
    const float sq = x2[gi] + x2j - 2.0f * c[v];
    float l = (sq > 0.0f) ? (-0.5f * sqrtf(sq)) : 0.0f;
    if (gi == gj) l = 0.0f;               // exact diagonal
    E[(size_t)gi * N + gj] = expf(l);     // row max is exactly 0 -> stable
    lsum += l;
  }

  __shared__ float red[256];
  red[threadIdx.x] = lsum;
  __syncthreads();
  for (int s = 128; s > 0; s >>= 1) {
    if (threadIdx.x < s) red[threadIdx.x] += red[threadIdx.x + s];
    __syncthreads();
  }
  if (threadIdx.x == 0)
    logitPart[blockIdx.y * gridDim.x + blockIdx.x] = red[0];
}

// ----------------------- per-row rank-masked log-sum (sort + suffix + search)
// blockIdx.x = row, blockIdx.y = 0 (theta) / 1 (shift)
__global__ __launch_bounds__(256) void rnc_rank(const float* __restrict__ theta,
                                                const float* __restrict__ shift,
                                                const float* __restrict__ E,
                                                float* __restrict__ logPart) {
  __shared__ float key[N];   // 16 KB
  __shared__ float val[N];   // 16 KB
  __shared__ float tmp[N];   // 16 KB (scan ping-pong)
  __shared__ float red[256];

  const int   row  = blockIdx.x;
  const int   tid  = threadIdx.x;
  const float* Drow = (blockIdx.y == 0 ? theta : shift) + (size_t)row * N;
  const float* Erow = E + (size_t)row * N;
  const float FINF  = __builtin_inff();

  // load; diagonal -> key=+inf, val=0 (equivalent to dropping it)
  for (int cidx = tid; cidx < N; cidx += 256) {
    const bool diag = (cidx == row);
    key[cidx] = diag ? FINF : Drow[cidx];
    val[cidx] = diag ? 0.0f : Erow[cidx];
  }
  __syncthreads();

  // bitonic sort ascending by key (val follows)
  for (int k = 2; k <= N; k <<= 1) {
    for (int j = k >> 1; j > 0; j >>= 1) {
      for (int i = tid; i < N; i += 256) {
        const int ixj = i ^ j;
        if (ixj > i) {
          const bool up = ((i & k) == 0);
          const float ki = key[i], kj = key[ixj];
          if (up ? (ki > kj) : (ki < kj)) {
            key[i] = kj; key[ixj] = ki;
            const float vi = val[i]; val[i] = val[ixj]; val[ixj] = vi;
          }
        }
      }
      __syncthreads();
    }
  }

  // inclusive suffix sum of sorted vals (Hillis-Steele, 12 passes -> ends in val)
  float* src = val;
  float* dst = tmp;
  for (int off = 1; off < N; off <<= 1) {
    for (int i = tid; i < N; i += 256)
      dst[i] = src[i] + ((i + off < N) ? src[i + off] : 0.0f);
    __syncthreads();
    float* t = src; src = dst; dst = t;
  }

  // searchsorted(left) per off-diagonal column + log accumulate
  float lsum = 0.0f;
  for (int cidx = tid; cidx < N; cidx += 256) {
    if (cidx == row) continue;
    const float v = Drow[cidx];
    int lo = 0, hi = N;
    while (lo < hi) {
      const int mid = (lo + hi) >> 1;
      if (key[mid] < v) lo = mid + 1; else hi = mid;
    }
    lsum += logf(src[lo]);
  }

  red[tid] = lsum;
  __syncthreads();
  for (int s = 128; s > 0; s >>= 1) {
    if (tid < s) red[tid] += red[tid + s];
    __syncthreads();
  }
  if (tid == 0) logPart[blockIdx.y * N + row] = red[0];
}

// ------------------------------------------------- deterministic final reduce
__global__ __launch_bounds__(256) void rnc_finalize(const float* __restrict__ logitPart,
                                                    int nLogit,
                                                    const float* __restrict__ logPart,
                                                    int nLog,
                                                    float* __restrict__ out) {
  __shared__ double red[256];
  const int tid = threadIdx.x;
  double s1 = 0.0, s2 = 0.0;
  for (int i = tid; i < nLogit; i += 256) s1 += (double)logitPart[i];
  for (int i = tid; i < nLog;  i += 256) s2 += (double)logPart[i];

  red[tid] = s1;
  __syncthreads();
  for (int s = 128; s > 0; s >>= 1) {
    if (tid < s) red[tid] += red[tid + s];
    __syncthreads();
  }
  const double sumLogits = red[0];
  __syncthreads();

  red[tid] = s2;
  __syncthreads();
  for (int s = 128; s > 0; s >>= 1) {
    if (tid < s) red[tid] += red[tid + s];
    __syncthreads();
  }
  const double sumLogs = red[0];

  if (tid == 0) {
    const double denom = (double)N * (double)(N - 1);
    // loss = (0.5*(S_theta + S_shift) - sum(logits_nd)) / (n*(n-1))
    out[0] = (float)((0.5 * sumLogs - sumLogits) / denom);
  }
}

extern "C" void kernel_launch(void* const* d_in, const int* in_sizes, int n_in,
                              void* d_out, int out_size, void* d_ws, size_t ws_size,
                              hipStream_t stream) {
  const float* theta = (const float*)d_in[0];
  const float* shift = (const float*)d_in[1];
  const float* F     = (const float*)d_in[2];
  float* out = (float*)d_out;
  float* ws  = (float*)d_ws;

  float* x2     = ws + WS_X2;
  float* logitP = ws + WS_LOGITP;
  float* logP   = ws + WS_LOGP;
  float* E      = ws + WS_E;

  rnc_rownorm<<<N, 128, 0, stream>>>(F, x2);

  dim3 gGram(32, 256);                       // 8 waves/block * 32 * 256 = 65536 tiles
  rnc_gram<<<gGram, 256, 0, stream>>>(F, x2, E, logitP);

  dim3 gRank(N, 2);                          // (row, theta/shift)
  rnc_rank<<<gRank, 256, 0, stream>>>(theta, shift, E, logP);

  rnc_finalize<<<1, 256, 0, stream>>>(logitP, 32 * 256, logP, 2 * N, out);
}